// log_Graph_Fokker_Planck_ODEFunc_678604833403
// MI455X (gfx1250) — compile-verified
//
#include <hip/hip_runtime.h>
#include <hip/hip_bf16.h>
#include <math.h>

// Problem constants (from reference)
#define B_   4
#define N_   512
#define D_   32
#define PE_  16

#define WPB 4           // waves (16-row i-tiles) per block
#define L2E 1.44269504088896340736f
#define INV_SQRT_D 0.17677669529663688110f   // 1/sqrt(32)
#define KS2 (10.0f * L2E)                    // K_SHARP in base-2 domain

typedef __attribute__((ext_vector_type(2))) float v2f;
typedef __attribute__((ext_vector_type(8))) float v8f;

static __device__ __forceinline__ v8f wmma_f32(v2f a, v2f b, v8f c) {
  // D = A(16x4,f32) x B(4x16,f32) + C(16x16,f32)
  return __builtin_amdgcn_wmma_f32_16x16x4_f32(false, a, false, b, (short)0, c, false, false);
}

static __device__ __forceinline__ float fexp2(float x) { return __builtin_amdgcn_exp2f(x); }
static __device__ __forceinline__ float frcp(float x)  { return __builtin_amdgcn_rcpf(x); }

// DPP16 cross-lane move (VALU, no LDS unit, no dscnt wait)
#define DPP_QUAD_XOR1  0xB1   // quad_perm [1,0,3,2]
#define DPP_QUAD_XOR2  0x4E   // quad_perm [2,3,0,1]
#define DPP_HALF_MIR   0x141  // row_half_mirror (swap quads within 8)
#define DPP_ROW_MIR    0x140  // row_mirror (swap 8-groups within 16)

template <int CTRL>
static __device__ __forceinline__ float dpp_mov(float x) {
  return __int_as_float(
      __builtin_amdgcn_update_dpp(0, __float_as_int(x), CTRL, 0xf, 0xf, true));
}
// Full reduction across each 16-lane row (result replicated in all 16 lanes)
static __device__ __forceinline__ float redmax16(float x) {
  x = fmaxf(x, dpp_mov<DPP_QUAD_XOR1>(x));
  x = fmaxf(x, dpp_mov<DPP_QUAD_XOR2>(x));
  x = fmaxf(x, dpp_mov<DPP_HALF_MIR>(x));
  x = fmaxf(x, dpp_mov<DPP_ROW_MIR>(x));
  return x;
}
static __device__ __forceinline__ float redsum16(float x) {
  x += dpp_mov<DPP_QUAD_XOR1>(x);
  x += dpp_mov<DPP_QUAD_XOR2>(x);
  x += dpp_mov<DPP_HALF_MIR>(x);
  x += dpp_mov<DPP_ROW_MIR>(x);
  return x;
}

// ---------------------------------------------------------------------------
// Kernel 1: K = pe @ Wk + bk ; Q = pe @ Wq + bq   (B*N x PE) @ (PE x D)
// ---------------------------------------------------------------------------
__global__ void kq_proj_kernel(const float* __restrict__ pe,
                               const float* __restrict__ Wk, const float* __restrict__ bk,
                               const float* __restrict__ Wq, const float* __restrict__ bq,
                               float* __restrict__ Kd, float* __restrict__ Qd) {
  int idx = blockIdx.x * blockDim.x + threadIdx.x;
  if (idx >= B_ * N_ * D_) return;
  int d  = idx & (D_ - 1);
  int bn = idx / D_;
  const float* p = pe + bn * PE_;
  float k = bk[d], q = bq[d];
#pragma unroll
  for (int e = 0; e < PE_; ++e) {
    float pv = p[e];
    k = fmaf(pv, Wk[e * D_ + d], k);
    q = fmaf(pv, Wq[e * D_ + d], q);
  }
  Kd[idx] = k;
  Qd[idx] = q;
}

// ---------------------------------------------------------------------------
// Kernel 2: fused flash-style Fokker-Planck ODE. One wave per 16-row i-tile.
// Softmax is computed entirely in the log2 domain (scale folded into K frags).
// ---------------------------------------------------------------------------
__global__ void __launch_bounds__(32 * WPB)
fokker_planck_fused_kernel(const float* __restrict__ h,
                           const float* __restrict__ E,
                           const float* __restrict__ A,
                           const float* __restrict__ beta,
                           const float* __restrict__ Kd,
                           const float* __restrict__ Qd,
                           float* __restrict__ out) {
  __shared__ float lds[WPB * 3 * 256];   // 3 x 16x16 f32 tiles per wave

  const int lane = threadIdx.x & 31;
  const int wid  = threadIdx.x >> 5;
  const int lo   = lane & 15;     // column / M within 16-lane half
  const int hi   = lane >> 4;     // half-wave select
  const int tile = blockIdx.x * WPB + wid;
  if (tile >= B_ * (N_ / 16)) return;
  const int b  = tile >> 5;             // tile / (N/16)
  const int i0 = (tile & 31) * 16;

  float* lds1 = lds + wid * 3 * 256;
  float* lds3 = lds1 + 256;
  float* lds4 = lds1 + 512;

  const float* Kb = Kd + b * N_ * D_;
  const float* Qb = Qd + b * N_ * D_;
  const float* Hb = h  + b * N_ * D_;

  // A-fragments of K_i (16x32), chunked K=4: lane holds row (i0+lo), k = 4*kc + 2*hi + {0,1}
  // 1/sqrt(D) * log2(e) folded in: scores come out as base-2 logits directly.
  const float kscale = INV_SQRT_D * L2E;
  v2f afK[8];
#pragma unroll
  for (int kc = 0; kc < 8; ++kc) {
    v2f t = *(const v2f*)(Kb + (i0 + lo) * D_ + 4 * kc + 2 * hi);
    afK[kc].x = t.x * kscale;
    afK[kc].y = t.y * kscale;
  }

  // E_i per fragment row (row = i0 + v + 8*hi)
  float Ei[8];
#pragma unroll
  for (int v = 0; v < 8; ++v) Ei[v] = E[i0 + v + 8 * hi];

  // online-softmax state + accumulators (all in log2 domain)
  v8f acc1[2] = {}, acc3[2] = {}, acc4[2] = {}, acc5[2] = {};
  float m[8], z[8], c2[8], c6[8];
#pragma unroll
  for (int v = 0; v < 8; ++v) { m[v] = -1.0e30f; z[v] = 0.f; c2[v] = 0.f; c6[v] = 0.f; }

  for (int j0 = 0; j0 < N_; j0 += 16) {
    // ---- scores tile S = (K_i*scale) @ Q_j^T : 8 chained f32 WMMAs over D=32 ----
    v8f sraw = {};
#pragma unroll
    for (int kc = 0; kc < 8; ++kc) {
      v2f bfQ = *(const v2f*)(Qb + (j0 + lo) * D_ + 4 * kc + 2 * hi);
      sraw = wmma_f32(afK[kc], bfQ, sraw);
    }

    const float Ej = E[j0 + lo];

    // graph mask (A is 0/1), per fragment element (row v+8*hi, col lo)
    float sc[8];
#pragma unroll
    for (int v = 0; v < 8; ++v)
      sc[v] = A[(i0 + v + 8 * hi) * N_ + j0 + lo] * sraw[v];

    // per-row tile max via DPP butterfly (stays within 16-lane rows)
    float tm[8];
#pragma unroll
    for (int v = 0; v < 8; ++v) tm[v] = redmax16(sc[v]);

    // rescale + exponentiate + build the three P-variants, stage to LDS
#pragma unroll
    for (int v = 0; v < 8; ++v) {
      float mn = fmaxf(m[v], tm[v]);
      float r  = fexp2(m[v] - mn);            // raw v_exp_f32, arg <= 0
      m[v] = mn;
      z[v]  *= r;  c2[v] *= r;  c6[v] *= r;
      acc1[0][v] *= r; acc1[1][v] *= r;
      acc3[0][v] *= r; acc3[1][v] *= r;
      acc4[0][v] *= r; acc4[1][v] *= r;
      acc5[0][v] *= r; acc5[1][v] *= r;

      float p  = fexp2(sc[v] - mn);           // unnormalized softmax weight
      float Ed = Ej - Ei[v];
      float sg = frcp(1.f + fexp2(-KS2 * Ed)); // sigmoid(10*Ed)
      float p3 = p * sg;          // P . s
      float p1 = p3 * Ed;         // P . Ed . s
      float p4 = p - p3;          // P . (1-s)
      z[v]  += p;
      c2[v] += p * Ed - p1;       // P . Ed . (1-s)
      c6[v] += p4;

      int addr = (v + 8 * hi) * 16 + lo;
      lds1[addr] = p1;
      lds3[addr] = p3;
      lds4[addr] = p4;
    }

    // C-fragment -> A-fragment layout conversion through LDS (in-order DS per wave)
    asm volatile("s_wait_dscnt 0" ::: "memory");

    v2f p1a[4], p3a[4], p4a[4];
#pragma unroll
    for (int kc = 0; kc < 4; ++kc) {
      int a = lo * 16 + 4 * kc + 2 * hi;
      p1a[kc] = *(const v2f*)(lds1 + a);
      p3a[kc] = *(const v2f*)(lds3 + a);
      p4a[kc] = *(const v2f*)(lds4 + a);
    }

    // value B-fragments: H_j, L_j, (L.H)_j  (B[k=j_local][n=d])
    v2f HB[4][2], LB[4][2], LHB[4][2];
#pragma unroll
    for (int kc = 0; kc < 4; ++kc)
#pragma unroll
      for (int s = 0; s < 2; ++s)
#pragma unroll
        for (int t = 0; t < 2; ++t) {
          float hv = Hb[(j0 + 4 * kc + 2 * hi + t) * D_ + lo + 16 * s];
          float lv = __logf(hv + 1e-8f);
          HB[kc][s][t]  = hv;
          LB[kc][s][t]  = lv;
          LHB[kc][s][t] = lv * hv;
        }

    // four value GEMMs: 4 x (4 chunks x 2 n-subtiles) WMMAs
#pragma unroll
    for (int s = 0; s < 2; ++s)
#pragma unroll
      for (int kc = 0; kc < 4; ++kc) {
        acc1[s] = wmma_f32(p1a[kc], HB[kc][s],  acc1[s]);
        acc5[s] = wmma_f32(p3a[kc], HB[kc][s],  acc5[s]);
        acc3[s] = wmma_f32(p3a[kc], LHB[kc][s], acc3[s]);
        acc4[s] = wmma_f32(p4a[kc], LB[kc][s],  acc4[s]);
      }
  }

  // ---- finalize: reduce row scalars across the 16-lane row (DPP) ----
#pragma unroll
  for (int v = 0; v < 8; ++v) {
    z[v]  = redsum16(z[v]);
    c2[v] = redsum16(c2[v]);
    c6[v] = redsum16(c6[v]);
  }

  float bt[2];
  bt[0] = beta[lo];
  bt[1] = beta[lo + 16];

#pragma unroll
  for (int v = 0; v < 8; ++v) {
    const int row = i0 + v + 8 * hi;
    const float invz = frcp(z[v]);
#pragma unroll
    for (int s = 0; s < 2; ++s) {
      const int col = lo + 16 * s;
      float hv = Hb[row * D_ + col];
      float lv = __logf(hv + 1e-8f);
      float num = acc1[s][v] + c2[v] * hv +
                  bt[s] * (acc3[s][v] + hv * acc4[s][v] - lv * acc5[s][v] - c6[v] * lv * hv);
      out[(b * N_ + row) * D_ + col] = num * invz;
    }
  }
}

// ---------------------------------------------------------------------------
extern "C" void kernel_launch(void* const* d_in, const int* in_sizes, int n_in,
                              void* d_out, int out_size, void* d_ws, size_t ws_size,
                              hipStream_t stream) {
  // inputs: 0:t 1:h 2:pe 3:E 4:A 5:Wk 6:bk 7:Wq 8:bq 9:beta
  const float* h    = (const float*)d_in[1];
  const float* pe   = (const float*)d_in[2];
  const float* E    = (const float*)d_in[3];
  const float* A    = (const float*)d_in[4];
  const float* Wk   = (const float*)d_in[5];
  const float* bk   = (const float*)d_in[6];
  const float* Wq   = (const float*)d_in[7];
  const float* bq   = (const float*)d_in[8];
  const float* beta = (const float*)d_in[9];
  float* out = (float*)d_out;

  float* Kd = (float*)d_ws;                       // B*N*D floats
  float* Qd = Kd + (size_t)B_ * N_ * D_;          // B*N*D floats

  {
    int total = B_ * N_ * D_;
    kq_proj_kernel<<<(total + 255) / 256, 256, 0, stream>>>(pe, Wk, bk, Wq, bq, Kd, Qd);
  }
  {
    int tiles = B_ * (N_ / 16);                   // 128 i-tiles, one wave each
    dim3 grid(tiles / WPB), block(32 * WPB);
    fokker_planck_fused_kernel<<<grid, block, 0, stream>>>(h, E, A, beta, Kd, Qd, out);
  }
}